// RWKV_RNN_36412732735592
// MI455X (gfx1250) — compile-verified
//
#include <hip/hip_runtime.h>

// RWKV-4 single-token inference for MI455X (gfx1250).
// Bandwidth-bound fp32 GEMV chain: ~1.5 GB weights/token -> ~65us HBM floor at
// 23.3 TB/s. All matvecs use V_WMMA_F32_16X16X4_F32 as the fp32 dot engine
// (A = activation slice broadcast from LDS, B = 16 distinct weight rows;
// K-step of 8 = 1 global_load_b128 + 1 ds_load_b128 + 2 wmma per lane).
// All elementwise stages (LN, token-shift mix, WKV recurrence, ln_out) are
// fused into the *prologue* of the consuming GEMV kernel: every block
// redundantly recomputes the 1024-wide vector (trivial vs. its MB-scale
// weight stream), eliminating ~100 one-block kernel drains. 98 launches total.

#define DD 1024
#define HH 4096
#define LL 24
#define VV 50277

typedef float v2f __attribute__((ext_vector_type(2)));
typedef float v8f __attribute__((ext_vector_type(8)));

// ---------------- block LayerNorm helper (256 threads, 4 elems/thread) -----
template <int BLK, int PER>
__device__ __forceinline__ void block_ln(const float (&in)[PER],
                                         const float* __restrict__ w,
                                         const float* __restrict__ b,
                                         float (&out)[PER], int n) {
  __shared__ float s_sum[BLK / 32];
  __shared__ float s_sq[BLK / 32];
  float sum = 0.f, sq = 0.f;
#pragma unroll
  for (int j = 0; j < PER; ++j) { sum += in[j]; sq += in[j] * in[j]; }
#pragma unroll
  for (int off = 16; off > 0; off >>= 1) {
    sum += __shfl_down(sum, off, 32);
    sq  += __shfl_down(sq, off, 32);
  }
  const int lane = threadIdx.x & 31, wv = threadIdx.x >> 5;
  if (lane == 0) { s_sum[wv] = sum; s_sq[wv] = sq; }
  __syncthreads();
  if (threadIdx.x == 0) {
    float S = 0.f, Q = 0.f;
    for (int i = 0; i < BLK / 32; ++i) { S += s_sum[i]; Q += s_sq[i]; }
    s_sum[0] = S; s_sq[0] = Q;
  }
  __syncthreads();
  const float mean = s_sum[0] / (float)n;
  const float var  = s_sq[0] / (float)n - mean * mean;
  const float inv  = rsqrtf(var + 1e-5f);
#pragma unroll
  for (int j = 0; j < PER; ++j) {
    const int i = threadIdx.x + j * BLK;
    out[j] = (in[j] - mean) * inv * w[i] + b[i];
  }
}

// ---------------- WMMA f32 GEMV row engine ---------------------------------
// y[r] = act( dot(W[r,:], xs) ), W row-major M x K (K compile-time, K%8==0).
// One wave owns 16 output rows; lane = 16*g + n. B operand: lane streams
// W[rowBase+n] as float4; A operand: xs slice broadcast from LDS. Result for
// row rowBase+n lands in C[0] of lanes 0..15.
// MODE: 0=identity 1=relu^2 3=sigmoid 4=base+dot 5=base+mul*dot
template <int MODE, int K>
__device__ __forceinline__ void gemv_rows(const float* __restrict__ W,
                                          const float* __restrict__ xs,
                                          float* __restrict__ y,
                                          const float* __restrict__ base,
                                          const float* __restrict__ mul,
                                          int M) {
  const int lane = threadIdx.x & 31;
  const int wv = threadIdx.x >> 5;
  const int wpb = blockDim.x >> 5;
  const int g = lane >> 4;   // K-half select
  const int n = lane & 15;   // output row within the 16-row group
  const float4* xs4 = (const float4*)xs;

  for (int rb = (blockIdx.x * wpb + wv) * 16; rb < M;
       rb += gridDim.x * wpb * 16) {
    int row = rb + n;
    if (row >= M) row = M - 1;  // clamp loads; store is predicated
    const float* wr = W + (size_t)row * K;

#if __has_builtin(__builtin_amdgcn_wmma_f32_16x16x4_f32)
    v8f c = {0.f, 0.f, 0.f, 0.f, 0.f, 0.f, 0.f, 0.f};
#pragma unroll 4
    for (int k0 = 0; k0 < K; k0 += 8) {
      const float4 wq = *(const float4*)(wr + k0 + 4 * g);  // global_load_b128
      const float4 aq = xs4[(k0 >> 2) + g];                 // ds_load_b128
      v2f a0 = {aq.x, aq.y}, b0 = {wq.x, wq.y};
      c = __builtin_amdgcn_wmma_f32_16x16x4_f32(false, a0, false, b0,
                                                (short)0, c, false, false);
      v2f a1 = {aq.z, aq.w}, b1 = {wq.z, wq.w};
      c = __builtin_amdgcn_wmma_f32_16x16x4_f32(false, a1, false, b1,
                                                (short)0, c, false, false);
    }
    const float dot = c[0];
#else
    float acc = 0.f;
#pragma unroll 4
    for (int k0 = 0; k0 < K; k0 += 8) {
      const float4 wq = *(const float4*)(wr + k0 + 4 * g);
      const float4 aq = xs4[(k0 >> 2) + g];
      acc += wq.x * aq.x + wq.y * aq.y + wq.z * aq.z + wq.w * aq.w;
    }
    const float dot = acc + __shfl_xor(acc, 16, 32);
#endif

    const int orow = rb + n;
    if (lane < 16 && orow < M) {
      float o;
      if constexpr (MODE == 0) {
        o = dot;
      } else if constexpr (MODE == 1) {
        const float t = fmaxf(dot, 0.f);
        o = t * t;
      } else if constexpr (MODE == 3) {
        o = 1.f / (1.f + expf(-dot));
      } else if constexpr (MODE == 4) {
        o = base[orow] + dot;
      } else {
        o = base[orow] + mul[orow] * dot;
      }
      y[orow] = o;
    }
  }
}

// ---------------- embedding + ln0 (one tiny kernel) ------------------------
__global__ __launch_bounds__(256) void k_embed_ln(const int* __restrict__ tok,
                                                  const float* __restrict__ emb,
                                                  const float* __restrict__ w,
                                                  const float* __restrict__ b,
                                                  float* __restrict__ xout) {
  const float* row = emb + (size_t)tok[0] * DD;
  float in[4], out[4];
#pragma unroll
  for (int j = 0; j < 4; ++j) in[j] = row[threadIdx.x + j * 256];
  block_ln<256, 4>(in, w, b, out, DD);
#pragma unroll
  for (int j = 0; j < 4; ++j) xout[threadIdx.x + j * 256] = out[j];
}

// ---------------- fused: LN1 + token-shift -> {kw,vw,rw} GEMVs -------------
// grid (8,3): blockIdx.y selects which matrix; each block recomputes LN+mix.
__global__ __launch_bounds__(256) void k_att3(
    const float* __restrict__ kw, const float* __restrict__ vw,
    const float* __restrict__ rw, const float* __restrict__ x,
    const float* __restrict__ st, const float* __restrict__ w,
    const float* __restrict__ b, const float* __restrict__ tmk,
    const float* __restrict__ tmv, const float* __restrict__ tmr,
    float* __restrict__ kb, float* __restrict__ vb, float* __restrict__ rb,
    float* __restrict__ nst) {
  __shared__ float xs[DD];
  float in[4], xn[4];
#pragma unroll
  for (int j = 0; j < 4; ++j) in[j] = x[threadIdx.x + j * 256];
  block_ln<256, 4>(in, w, b, xn, DD);
  const float* tm = (blockIdx.y == 0) ? tmk : (blockIdx.y == 1) ? tmv : tmr;
#pragma unroll
  for (int j = 0; j < 4; ++j) {
    const int i = threadIdx.x + j * 256;
    const float s1 = st[DD + i];
    const float t = tm[i];
    xs[i] = xn[j] * t + s1 * (1.f - t);
    if (blockIdx.x == 0 && blockIdx.y == 0) nst[DD + i] = xn[j];  // new_s[1]
  }
  __syncthreads();
  if (blockIdx.y == 0)      gemv_rows<0, DD>(kw, xs, kb, nullptr, nullptr, DD);
  else if (blockIdx.y == 1) gemv_rows<0, DD>(vw, xs, vb, nullptr, nullptr, DD);
  else                      gemv_rows<0, DD>(rw, xs, rb, nullptr, nullptr, DD);
}

// ---------------- fused: sigmoid + WKV recurrence -> ow GEMV (x += ...) ----
__global__ __launch_bounds__(256) void k_wkv_ow(
    const float* __restrict__ ow, const float* __restrict__ kb,
    const float* __restrict__ vb, const float* __restrict__ rb,
    const float* __restrict__ st, const float* __restrict__ tf,
    const float* __restrict__ td, float* __restrict__ x,
    float* __restrict__ nst) {
  __shared__ float xs[DD];  // rwkv = sigmoid(r) * wkv
#pragma unroll
  for (int j = 0; j < 4; ++j) {
    const int i = threadIdx.x + j * 256;
    const float k = kb[i], v = vb[i];
    const float r = 1.f / (1.f + expf(-rb[i]));
    const float aa = st[2 * DD + i], bb = st[3 * DD + i], pp = st[4 * DD + i];
    const float ww = tf[i] + k;
    const float qq = fmaxf(pp, ww);
    const float e1 = expf(pp - qq), e2 = expf(ww - qq);
    const float wkv = (e1 * aa + e2 * v) / (e1 * bb + e2);
    const float ww2 = pp + td[i];
    const float qq2 = fmaxf(ww2, k);
    const float e1b = expf(ww2 - qq2), e2b = expf(k - qq2);
    if (blockIdx.x == 0) {
      nst[2 * DD + i] = e1b * aa + e2b * v;  // aa_n
      nst[3 * DD + i] = e1b * bb + e2b;      // bb_n
      nst[4 * DD + i] = qq2;                 // pp_n
    }
    xs[i] = r * wkv;
  }
  __syncthreads();
  gemv_rows<4, DD>(ow, xs, x, x, nullptr, DD);  // x += ow @ rwkv
}

// ---------------- fused: LN2 + token-shift -> {frw sigmoid, fkw relu^2} ----
// grid (32,2): y=0 -> frw (M=DD, only blocks 0..7 have rows), y=1 -> fkw.
__global__ __launch_bounds__(256) void k_ffn2(
    const float* __restrict__ frw, const float* __restrict__ fkw,
    const float* __restrict__ x, const float* __restrict__ st,
    const float* __restrict__ w, const float* __restrict__ b,
    const float* __restrict__ fmk, const float* __restrict__ fmr,
    float* __restrict__ frb, float* __restrict__ fkb,
    float* __restrict__ nst) {
  __shared__ float xs[DD];
  float in[4], xn[4];
#pragma unroll
  for (int j = 0; j < 4; ++j) in[j] = x[threadIdx.x + j * 256];
  block_ln<256, 4>(in, w, b, xn, DD);
  const float* tm = (blockIdx.y == 0) ? fmr : fmk;
#pragma unroll
  for (int j = 0; j < 4; ++j) {
    const int i = threadIdx.x + j * 256;
    const float s0 = st[i];
    const float t = tm[i];
    xs[i] = xn[j] * t + s0 * (1.f - t);
    if (blockIdx.x == 0 && blockIdx.y == 0) nst[i] = xn[j];  // new_s[0] = xn2
  }
  __syncthreads();
  if (blockIdx.y == 0) gemv_rows<3, DD>(frw, xs, frb, nullptr, nullptr, DD);
  else                 gemv_rows<1, DD>(fkw, xs, fkb, nullptr, nullptr, HH);
}

// ---------------- fused: stage fk -> x += fr * (fvw @ fk) ------------------
__global__ __launch_bounds__(256) void k_fv(const float* __restrict__ fvw,
                                            const float* __restrict__ fkb,
                                            const float* __restrict__ frb,
                                            float* __restrict__ x) {
  __shared__ float xs[HH];
  for (int i = threadIdx.x; i < HH; i += 256) xs[i] = fkb[i];
  __syncthreads();
  gemv_rows<5, HH>(fvw, xs, x, x, frb, DD);
}

// ---------------- fused: ln_out -> head GEMV -------------------------------
__global__ __launch_bounds__(256) void k_head(const float* __restrict__ hw,
                                              const float* __restrict__ x,
                                              const float* __restrict__ w,
                                              const float* __restrict__ b,
                                              float* __restrict__ logits) {
  __shared__ float xs[DD];
  float in[4], out[4];
#pragma unroll
  for (int j = 0; j < 4; ++j) in[j] = x[threadIdx.x + j * 256];
  block_ln<256, 4>(in, w, b, out, DD);
#pragma unroll
  for (int j = 0; j < 4; ++j) xs[threadIdx.x + j * 256] = out[j];
  __syncthreads();
  gemv_rows<0, DD>(hw, xs, logits, nullptr, nullptr, VV);
}

// ---------------- driver ----------------------------------------------------
extern "C" void kernel_launch(void* const* d_in, const int* in_sizes, int n_in,
                              void* d_out, int out_size, void* d_ws,
                              size_t ws_size, hipStream_t stream) {
  const int*   token   = (const int*)d_in[0];
  const float* state   = (const float*)d_in[1];
  const float* emb     = (const float*)d_in[2];
  const float* ln0_w   = (const float*)d_in[3];
  const float* ln0_b   = (const float*)d_in[4];
  const float* ln1_w   = (const float*)d_in[5];
  const float* ln1_b   = (const float*)d_in[6];
  const float* ln2_w   = (const float*)d_in[7];
  const float* ln2_b   = (const float*)d_in[8];
  const float* tmk     = (const float*)d_in[9];
  const float* tmv     = (const float*)d_in[10];
  const float* tmr     = (const float*)d_in[11];
  const float* tf      = (const float*)d_in[12];
  const float* td      = (const float*)d_in[13];
  const float* kw      = (const float*)d_in[14];
  const float* vw      = (const float*)d_in[15];
  const float* rw      = (const float*)d_in[16];
  const float* ow      = (const float*)d_in[17];
  const float* fmk     = (const float*)d_in[18];
  const float* fmr     = (const float*)d_in[19];
  const float* fkw     = (const float*)d_in[20];
  const float* fvw     = (const float*)d_in[21];
  const float* frw     = (const float*)d_in[22];
  const float* lnout_w = (const float*)d_in[23];
  const float* lnout_b = (const float*)d_in[24];
  const float* head_w  = (const float*)d_in[25];

  float* wsf  = (float*)d_ws;
  float* x    = wsf + 0;
  float* kb   = wsf + 1 * DD;
  float* vb   = wsf + 2 * DD;
  float* rb   = wsf + 3 * DD;
  float* frb  = wsf + 4 * DD;
  float* fkb  = wsf + 5 * DD;  // HH floats; total 9216 floats = 36 KB

  float* logits = (float*)d_out;
  float* nstate = (float*)d_out + VV;

  k_embed_ln<<<1, 256, 0, stream>>>(token, emb, ln0_w, ln0_b, x);

  for (int l = 0; l < LL; ++l) {
    const float* st = state + (size_t)l * 5 * DD;
    float* nst = nstate + (size_t)l * 5 * DD;

    k_att3<<<dim3(8, 3), 256, 0, stream>>>(
        kw + (size_t)l * DD * DD, vw + (size_t)l * DD * DD,
        rw + (size_t)l * DD * DD, x, st, ln1_w + l * DD, ln1_b + l * DD,
        tmk + l * DD, tmv + l * DD, tmr + l * DD, kb, vb, rb, nst);

    k_wkv_ow<<<8, 256, 0, stream>>>(ow + (size_t)l * DD * DD, kb, vb, rb, st,
                                    tf + l * DD, td + l * DD, x, nst);

    k_ffn2<<<dim3(32, 2), 256, 0, stream>>>(
        frw + (size_t)l * DD * DD, fkw + (size_t)l * HH * DD, x, st,
        ln2_w + l * DD, ln2_b + l * DD, fmk + l * DD, fmr + l * DD, frb, fkb,
        nst);

    k_fv<<<8, 256, 0, stream>>>(fvw + (size_t)l * DD * HH, fkb, frb, x);
  }

  k_head<<<((VV + 15) / 16 + 7) / 8, 256, 0, stream>>>(head_w, x, lnout_w,
                                                       lnout_b, logits);
}